// RaftAttentionLayer_62457414419129
// MI455X (gfx1250) — compile-verified
//
#include <hip/hip_runtime.h>
#include <math.h>
#include <stdint.h>

typedef __attribute__((ext_vector_type(2))) float v2f;
typedef __attribute__((ext_vector_type(8))) float v8f;

#define B_   2
#define S_   2048
#define H_   1024
#define NH_  16
#define HD_  64
#define NIDX 50000
#define M_   (B_*S_)                     // 4096 query rows
#define JSPLIT 32
#define KEYS_PER ((NIDX + JSPLIT - 1) / JSPLIT)   // 1563

#define BK 64
#define SK 68    // LDS row stride (floats): 68 % 64 = 4 -> conflict-free b64 lanes

__device__ __forceinline__ v8f wmma4(v2f a, v2f b, v8f c) {
  // V_WMMA_F32_16X16X4_F32 : D = A(16x4,f32) x B(4x16,f32) + C(16x16,f32)
  return __builtin_amdgcn_wmma_f32_16x16x4_f32(false, a, false, b, (short)0, c,
                                               false, false);
}

// Async DMA: 16 bytes per lane, global -> LDS, tracked by ASYNCcnt (in-order).
__device__ __forceinline__ void async_cp16(void* lds, const void* gptr) {
  asm volatile("global_load_async_to_lds_b128 %0, %1, off"
               :: "v"((uint32_t)(uintptr_t)lds),
                  "v"((uint64_t)(uintptr_t)gptr)
               : "memory");
}
template <int N>
__device__ __forceinline__ void async_wait_le() {
  asm volatile("s_wait_asynccnt %0" :: "i"(N) : "memory");
}

// ---------------------------------------------------------------------------
// ||k||^2 per index key row. One wave per row.
// ---------------------------------------------------------------------------
__global__ __launch_bounds__(256) void keynorm_kernel(
    const float* __restrict__ keys, float* __restrict__ knorm) {
  const int wave = threadIdx.x >> 5;
  const int lane = threadIdx.x & 31;
  const int row  = blockIdx.x * 8 + wave;
  const float* kp = keys + (size_t)row * H_;
  float s = 0.f;
#pragma unroll
  for (int i = 0; i < H_ / 32; ++i) {
    float x = kp[lane + 32 * i];
    s += x * x;
  }
#pragma unroll
  for (int off = 16; off >= 1; off >>= 1) s += __shfl_xor(s, off, 32);
  if (lane == 0) knorm[row] = s;
}

// ---------------------------------------------------------------------------
// C[m,n] = sum_k A[m,k] * W[n,k] + bias[n]   (NT GEMM, K = H_ = 1024)
// Block tile 64x64, K-tile 64, double-buffered async-DMA pipeline.
// 8 waves; each wave: one 16-row A-frag reused across two 16x16 C tiles.
// ---------------------------------------------------------------------------
__global__ __launch_bounds__(256) void gemm_nt_kernel(
    const float* __restrict__ A, const float* __restrict__ W,
    const float* __restrict__ bias, float* __restrict__ C, int N) {
  __shared__ __align__(16) float As[2][64 * SK];
  __shared__ __align__(16) float Bs[2][64 * SK];
  const int tid  = threadIdx.x;
  const int wave = tid >> 5, lane = tid & 31;
  const int half = lane >> 4, l16 = lane & 15;
  const int mt = wave >> 1;
  const int nt0 = (wave & 1) * 2, nt1 = nt0 + 1;
  const int m0 = blockIdx.x * 64;
  const int n0 = blockIdx.y * 64;
  const int NB = H_ / BK;                      // 16 k-blocks

  auto stage = [&](int buf, int kb) {          // 8 async DMA instrs / wave
#pragma unroll
    for (int t = 0; t < 4; ++t) {              // A tile: 64 x 64
      int id = tid + t * 256;
      int row = id >> 4, c4 = (id & 15) << 2;
      async_cp16(&As[buf][row * SK + c4], &A[(size_t)(m0 + row) * H_ + kb + c4]);
    }
#pragma unroll
    for (int t = 0; t < 4; ++t) {              // W tile: 64 x 64
      int id = tid + t * 256;
      int row = id >> 4, c4 = (id & 15) << 2;
      async_cp16(&Bs[buf][row * SK + c4], &W[(size_t)(n0 + row) * H_ + kb + c4]);
    }
  };

  v8f acc0 = {}, acc1 = {};
  stage(0, 0);
  for (int i = 0; i < NB; ++i) {
    const int cur = i & 1;
    if (i + 1 < NB) { stage(cur ^ 1, (i + 1) * BK); async_wait_le<8>(); }
    else            { async_wait_le<0>(); }
    __syncthreads();                           // current buffer ready (all waves)
    const float* ap  = &As[cur][(mt * 16 + l16) * SK + 2 * half];
    const float* bp0 = &Bs[cur][(nt0 * 16 + l16) * SK + 2 * half];
    const float* bp1 = &Bs[cur][(nt1 * 16 + l16) * SK + 2 * half];
#pragma unroll
    for (int kk = 0; kk < BK; kk += 4) {
      v2f a = *(const v2f*)(ap + kk);
      acc0 = wmma4(a, *(const v2f*)(bp0 + kk), acc0);
      acc1 = wmma4(a, *(const v2f*)(bp1 + kk), acc1);
    }
    __syncthreads();                           // buffer reusable for i+2 stage
  }

  const int na = n0 + nt0 * 16 + l16;
  const int nb = n0 + nt1 * 16 + l16;
  const float ba = bias ? bias[na] : 0.f;
  const float bb = bias ? bias[nb] : 0.f;
#pragma unroll
  for (int r = 0; r < 8; ++r) {
    int m = m0 + mt * 16 + 8 * half + r;       // C layout: VGPR r -> row r / r+8
    C[(size_t)m * N + na] = acc0[r] + ba;
    C[(size_t)m * N + nb] = acc1[r] + bb;
  }
}

// ---------------------------------------------------------------------------
// kNN: scores = -2 q.k + ||k||^2, running argmin over this block's key range.
// Grid (M/64, JSPLIT); 64-key chunks; double-buffered async-DMA pipeline.
// ---------------------------------------------------------------------------
__global__ __launch_bounds__(256) void knn_kernel(
    const float* __restrict__ Q, const float* __restrict__ keys,
    const float* __restrict__ knorm, float* __restrict__ pval,
    int* __restrict__ pidx) {
  __shared__ __align__(16) float As[2][64 * SK];
  __shared__ __align__(16) float Bs[2][64 * SK];
  __shared__ float lbv[2][64];
  __shared__ int   lbi[2][64];
  const int tid  = threadIdx.x;
  const int wave = tid >> 5, lane = tid & 31;
  const int half = lane >> 4, l16 = lane & 15;
  const int mt = wave >> 1;
  const int nt0 = (wave & 1) * 2, nt1 = nt0 + 1;
  const int m0 = blockIdx.x * 64;
  const int j  = blockIdx.y;
  const int rs = j * KEYS_PER;
  const int re = min(rs + KEYS_PER, NIDX);
  const int nchunks = (re - rs + 63) >> 6;
  const int NB = H_ / BK;                      // 16 k-blocks

  float bestv[8];
  int   besti[8];
#pragma unroll
  for (int r = 0; r < 8; ++r) { bestv[r] = INFINITY; besti[r] = 0; }

  for (int ch = 0; ch < nchunks; ++ch) {
    const int n0c = rs + ch * 64;

    auto stage = [&](int buf, int kb) {        // 8 async DMA instrs / wave
#pragma unroll
      for (int t = 0; t < 4; ++t) {            // Q tile 64 x 64
        int id = tid + t * 256;
        int row = id >> 4, c4 = (id & 15) << 2;
        async_cp16(&As[buf][row * SK + c4],
                   &Q[(size_t)(m0 + row) * H_ + kb + c4]);
      }
#pragma unroll
      for (int t = 0; t < 4; ++t) {            // key tile 64 x 64 (clamped)
        int id = tid + t * 256;
        int row = id >> 4, c4 = (id & 15) << 2;
        int n = n0c + row;
        if (n >= NIDX) n = 0;                  // OOB rows masked in epilogue
        async_cp16(&Bs[buf][row * SK + c4],
                   &keys[(size_t)n * H_ + kb + c4]);
      }
    };

    v8f acc0 = {}, acc1 = {};
    stage(0, 0);
    for (int i = 0; i < NB; ++i) {
      const int cur = i & 1;
      if (i + 1 < NB) { stage(cur ^ 1, (i + 1) * BK); async_wait_le<8>(); }
      else            { async_wait_le<0>(); }
      __syncthreads();
      const float* ap  = &As[cur][(mt * 16 + l16) * SK + 2 * half];
      const float* bp0 = &Bs[cur][(nt0 * 16 + l16) * SK + 2 * half];
      const float* bp1 = &Bs[cur][(nt1 * 16 + l16) * SK + 2 * half];
#pragma unroll
      for (int kk = 0; kk < BK; kk += 4) {
        v2f a = *(const v2f*)(ap + kk);
        acc0 = wmma4(a, *(const v2f*)(bp0 + kk), acc0);
        acc1 = wmma4(a, *(const v2f*)(bp1 + kk), acc1);
      }
      __syncthreads();
    }

    const int ng0 = n0c + nt0 * 16 + l16;
    const int ng1 = ng0 + 16;
    const bool va0 = ng0 < re, va1 = ng1 < re;
    const float kn0 = va0 ? knorm[ng0] : 0.f;
    const float kn1 = va1 ? knorm[ng1] : 0.f;
#pragma unroll
    for (int r = 0; r < 8; ++r) {
      float sc = va0 ? (-2.f * acc0[r] + kn0) : INFINITY;
      int   ii = ng0;
#pragma unroll
      for (int off = 8; off >= 1; off >>= 1) {   // 16-lane halves = C rows
        float ov = __shfl_xor(sc, off, 32);
        int   oi = __shfl_xor(ii, off, 32);
        if (ov < sc || (ov == sc && oi < ii)) { sc = ov; ii = oi; }
      }
      if (sc < bestv[r]) { bestv[r] = sc; besti[r] = ii; }

      float sd = va1 ? (-2.f * acc1[r] + kn1) : INFINITY;
      int   id1 = ng1;
#pragma unroll
      for (int off = 8; off >= 1; off >>= 1) {
        float ov = __shfl_xor(sd, off, 32);
        int   oi = __shfl_xor(id1, off, 32);
        if (ov < sd || (ov == sd && oi < id1)) { sd = ov; id1 = oi; }
      }
      if (sd < bestv[r]) { bestv[r] = sd; besti[r] = id1; }
    }
  }

  if (l16 == 0) {
#pragma unroll
    for (int r = 0; r < 8; ++r) {
      int rl = mt * 16 + 8 * half + r;
      lbv[wave & 1][rl] = bestv[r];
      lbi[wave & 1][rl] = besti[r];
    }
  }
  __syncthreads();
  if (tid < 64) {
    float v0 = lbv[0][tid], v1 = lbv[1][tid];
    int   i0 = lbi[0][tid], i1 = lbi[1][tid];
    bool take1 = (v1 < v0) || (v1 == v0 && i1 < i0);
    pval[(size_t)j * M_ + m0 + tid] = take1 ? v1 : v0;
    pidx[(size_t)j * M_ + m0 + tid] = take1 ? i1 : i0;
  }
}

__global__ __launch_bounds__(256) void knn_reduce_kernel(
    const float* __restrict__ pval, const int* __restrict__ pidx,
    int* __restrict__ idxf) {
  const int row = blockIdx.x * 256 + threadIdx.x;
  float best = INFINITY;
  int bidx = 0;
  for (int j = 0; j < JSPLIT; ++j) {
    float v = pval[(size_t)j * M_ + row];
    int   i = pidx[(size_t)j * M_ + row];
    if (v < best || (v == best && i < bidx)) { best = v; bidx = i; }
  }
  idxf[row] = bidx;
}

__global__ __launch_bounds__(256) void gather_kernel(
    const float* __restrict__ vals, const int* __restrict__ idxf,
    float* __restrict__ out) {
  const int row = blockIdx.x;
  const int idx = idxf[row];
  const float4* src = (const float4*)(vals + (size_t)idx * H_);
  float4* dst = (float4*)(out + (size_t)row * H_);
  dst[threadIdx.x] = src[threadIdx.x];
}

// ---------------------------------------------------------------------------
// Flash attention: per (q-block=64, head, batch). Key chunks of 32.
// ---------------------------------------------------------------------------
#define SQ 68
#define SP 36

__global__ __launch_bounds__(256) void flash_kernel(
    const float* __restrict__ Q, const float* __restrict__ K,
    const float* __restrict__ V, float* __restrict__ O) {
  __shared__ __align__(16) float Qs[64 * SQ];
  __shared__ __align__(16) float Ks[32 * SQ];
  __shared__ float Vst[64 * SP];          // V transposed: [d][kc]
  __shared__ float Ps[64 * SP];           // P tile in A-consumable layout
  __shared__ float mrow[64], lrow[64], alphas[64];
  __shared__ float tmax[2][64], tsum[2][64];

  const int tid  = threadIdx.x;
  const int wave = tid >> 5, lane = tid & 31;
  const int half = lane >> 4, l16 = lane & 15;
  const int mt = wave >> 1, nt = wave & 1;
  const int qb = blockIdx.x, h = blockIdx.y, b = blockIdx.z;
  const size_t base = (size_t)b * S_ * H_ + (size_t)h * HD_;
  const int q0 = qb * 64;

#pragma unroll
  for (int t = 0; t < 4; ++t) {           // Q tile 64 x 64 (async DMA)
    int id = tid + t * 256;
    int row = id >> 4, d4 = (id & 15) << 2;
    async_cp16(&Qs[row * SQ + d4], &Q[base + (size_t)(q0 + row) * H_ + d4]);
  }
  if (tid < 64) { mrow[tid] = -INFINITY; lrow[tid] = 0.f; }

  v8f acc0 = {}, acc1 = {};
  const int nt0 = 2 * nt, nt1 = 2 * nt + 1;

  for (int kb = 0; kb < S_ / 32; ++kb) {
    __syncthreads();
#pragma unroll
    for (int t = 0; t < 2; ++t) {         // K tile async; V transposed via regs
      int id = tid + t * 256;
      int row = id >> 4, d4 = (id & 15) << 2;
      async_cp16(&Ks[row * SQ + d4],
                 &K[base + (size_t)(kb * 32 + row) * H_ + d4]);
      float4 vv = *(const float4*)&V[base + (size_t)(kb * 32 + row) * H_ + d4];
      Vst[(d4 + 0) * SP + row] = vv.x;
      Vst[(d4 + 1) * SP + row] = vv.y;
      Vst[(d4 + 2) * SP + row] = vv.z;
      Vst[(d4 + 3) * SP + row] = vv.w;
    }
    async_wait_le<0>();
    __syncthreads();

    // S = Q Kt / sqrt(HD): one 16x16 tile per wave of the 64x32 block
    v8f s = {};
    {
      const float* ap = &Qs[(mt * 16 + l16) * SQ + 2 * half];
      const float* bp = &Ks[(nt * 16 + l16) * SQ + 2 * half];
#pragma unroll
      for (int kk = 0; kk < HD_; kk += 4)
        s = wmma4(*(const v2f*)(ap + kk), *(const v2f*)(bp + kk), s);
    }

    float rmax[8];
#pragma unroll
    for (int r = 0; r < 8; ++r) {
      s[r] *= 0.125f;
      float v = s[r];
#pragma unroll
      for (int off = 8; off >= 1; off >>= 1) v = fmaxf(v, __shfl_xor(v, off, 32));
      rmax[r] = v;
    }
    if (l16 == 0) {
#pragma unroll
      for (int r = 0; r < 8; ++r) tmax[nt][mt * 16 + 8 * half + r] = rmax[r];
    }
    __syncthreads();
    if (tid < 64) {
      float mo = mrow[tid];
      float mn = fmaxf(mo, fmaxf(tmax[0][tid], tmax[1][tid]));
      mrow[tid]   = mn;
      alphas[tid] = __expf(mo - mn);
    }
    __syncthreads();

    float rsum[8];
#pragma unroll
    for (int r = 0; r < 8; ++r) {
      int row = mt * 16 + 8 * half + r;
      float p = __expf(s[r] - mrow[row]);
      Ps[row * SP + nt * 16 + l16] = p;
      float t = p;
#pragma unroll
      for (int off = 8; off >= 1; off >>= 1) t += __shfl_xor(t, off, 32);
      rsum[r] = t;
    }
    if (l16 == 0) {
#pragma unroll
      for (int r = 0; r < 8; ++r) tsum[nt][mt * 16 + 8 * half + r] = rsum[r];
    }
    __syncthreads();
    if (tid < 64) lrow[tid] = lrow[tid] * alphas[tid] + tsum[0][tid] + tsum[1][tid];

#pragma unroll
    for (int r = 0; r < 8; ++r) {
      float a = alphas[mt * 16 + 8 * half + r];
      acc0[r] *= a;
      acc1[r] *= a;
    }
    // acc += P @ V  (contraction over the 32-key chunk)
    {
      const float* ap = &Ps[(mt * 16 + l16) * SP + 2 * half];
      const float* b0 = &Vst[(nt0 * 16 + l16) * SP + 2 * half];
      const float* b1 = &Vst[(nt1 * 16 + l16) * SP + 2 * half];
#pragma unroll
      for (int kk = 0; kk < 32; kk += 4) {
        v2f a = *(const v2f*)(ap + kk);
        acc0 = wmma4(a, *(const v2f*)(b0 + kk), acc0);
        acc1 = wmma4(a, *(const v2f*)(b1 + kk), acc1);
      }
    }
  }

  __syncthreads();
#pragma unroll
  for (int r = 0; r < 8; ++r) {
    int row = mt * 16 + 8 * half + r;
    float inv = 1.f / lrow[row];
    size_t o = base + (size_t)(q0 + row) * H_;
    O[o + nt0 * 16 + l16] = acc0[r] * inv;
    O[o + nt1 * 16 + l16] = acc1[r] * inv;
  }
}

// ---------------------------------------------------------------------------
extern "C" void kernel_launch(void* const* d_in, const int* in_sizes, int n_in,
                              void* d_out, int out_size, void* d_ws,
                              size_t ws_size, hipStream_t stream) {
  (void)in_sizes; (void)n_in; (void)out_size; (void)ws_size;
  const float* hidden = (const float*)d_in[0];
  const float* Wq = (const float*)d_in[1];
  const float* bq = (const float*)d_in[2];
  const float* Wk = (const float*)d_in[3];
  const float* bk = (const float*)d_in[4];
  const float* Wv = (const float*)d_in[5];
  const float* bv = (const float*)d_in[6];
  const float* Wo = (const float*)d_in[7];
  const float* bo = (const float*)d_in[8];
  const float* ikeys = (const float*)d_in[9];
  const float* ivals = (const float*)d_in[10];
  float* out = (float*)d_out;

  const size_t MH = (size_t)M_ * H_;
  float* ws        = (float*)d_ws;
  float* q         = ws;
  float* retrieved = q + MH;
  float* kmat      = retrieved + MH;
  float* vmat      = kmat + MH;
  float* ctx       = vmat + MH;
  float* knorm     = ctx + MH;                  // 50000 (padded)
  float* pval      = knorm + 50048;             // JSPLIT * M_
  int*   pidx      = (int*)(pval + (size_t)JSPLIT * M_);
  int*   idxf      = pidx + (size_t)JSPLIT * M_;

  keynorm_kernel<<<NIDX / 8, 256, 0, stream>>>(ikeys, knorm);
  gemm_nt_kernel<<<dim3(M_ / 64, H_ / 64), 256, 0, stream>>>(hidden, Wq, bq, q, H_);
  knn_kernel<<<dim3(M_ / 64, JSPLIT), 256, 0, stream>>>(q, ikeys, knorm, pval, pidx);
  knn_reduce_kernel<<<M_ / 256, 256, 0, stream>>>(pval, pidx, idxf);
  gather_kernel<<<M_, 256, 0, stream>>>(ivals, idxf, retrieved);
  gemm_nt_kernel<<<dim3(M_ / 64, H_ / 64), 256, 0, stream>>>(retrieved, Wk, bk, kmat, H_);
  gemm_nt_kernel<<<dim3(M_ / 64, H_ / 64), 256, 0, stream>>>(retrieved, Wv, bv, vmat, H_);
  flash_kernel<<<dim3(S_ / 64, NH_, B_), 256, 0, stream>>>(q, kmat, vmat, ctx);
  gemm_nt_kernel<<<dim3(M_ / 64, H_ / 64), 256, 0, stream>>>(ctx, Wo, bo, out, H_);
}